// DCAPLayer_4604204941553
// MI455X (gfx1250) — compile-verified
//
#include <hip/hip_runtime.h>
#include <hip/hip_bf16.h>

#define BATCH   4096
#define F       100          // FIELDS
#define FP      112          // FIELDS padded to 7 tiles of 16
#define DD      64           // D
#define NPAIR   4950         // F*(F-1)/2
#define XS      68           // LDS stride for X rows (16B aligned, conflict-light)
#define QS      17           // LDS stride for Q/K (coprime with 64 banks)

typedef float v2f __attribute__((ext_vector_type(2)));
typedef float v8f __attribute__((ext_vector_type(8)));
typedef int   v4i __attribute__((__vector_size__(16)));

#if defined(__has_builtin)
#if __has_builtin(__builtin_amdgcn_global_load_async_to_lds_b128)
#define HAVE_ASYNC_LDS 1
#endif
#endif

__device__ __forceinline__ v8f wmma_f32(v2f a, v2f b, v8f c) {
    // D = A(16x4,f32) * B(4x16,f32) + C(16x16,f32)
    return __builtin_amdgcn_wmma_f32_16x16x4_f32(false, a, false, b,
                                                 (short)0, c, false, false);
}

#ifdef HAVE_ASYNC_LDS
__device__ __forceinline__ void async_g2l_b128(const float* g, float* l) {
    __builtin_amdgcn_global_load_async_to_lds_b128(
        (__attribute__((address_space(1))) v4i*)(g),
        (__attribute__((address_space(3))) v4i*)(l), 0, 0);
}
__device__ __forceinline__ void async_wait0() {
#if __has_builtin(__builtin_amdgcn_s_wait_asynccnt)
    __builtin_amdgcn_s_wait_asynccnt(0);
#else
    asm volatile("s_wait_asynccnt 0x0" ::: "memory");
#endif
}
#endif

__launch_bounds__(128, 1)
__global__ void dcap_attn_pairs_kernel(const float* __restrict__ x,
                                       const float* __restrict__ wq,
                                       const float* __restrict__ wk,
                                       const float* __restrict__ wv,
                                       float* __restrict__ out) {
    __shared__ float sX[FP * XS];    // X_b, padded rows zeroed
    __shared__ float sWq[DD * 16];   // wq[:, 0:16]
    __shared__ float sWk[DD * 16];   // wk[:, 0:16]
    __shared__ float sWv[DD];        // wv[:, 0]
    __shared__ float sQ[FP * QS];    // Q = X*Wq0
    __shared__ float sK[FP * QS];    // K = X*Wk0
    __shared__ float sVcol[FP];      // v column (head0, dim0)
    __shared__ float sZ[FP];         // z[:, 0]

    const int b    = blockIdx.x;
    const int tid  = threadIdx.x;
    const int wave = __builtin_amdgcn_readfirstlane(tid >> 5);  // sgpr-uniform
    const int lane = tid & 31;
    const int half = lane >> 4;      // 0: lanes 0-15, 1: lanes 16-31
    const int lrow = lane & 15;

    // ---- stage inputs into LDS --------------------------------------------
    {
        const float* xb = x + (size_t)b * F * DD;
#ifdef HAVE_ASYNC_LDS
        // 100 rows x 64 floats = 1600 x 16B chunks, copied via ASYNCcnt path
        for (int idx = tid; idx < F * (DD / 4); idx += 128) {
            int row = idx >> 4, c4 = idx & 15;
            async_g2l_b128(xb + row * DD + c4 * 4, &sX[row * XS + c4 * 4]);
        }
#else
        const float4* xg = reinterpret_cast<const float4*>(xb);
        for (int idx = tid; idx < F * (DD / 4); idx += 128) {
            int row = idx >> 4, c4 = idx & 15;
            float4 v = xg[row * (DD / 4) + c4];
            float* dst = &sX[row * XS + c4 * 4];
            dst[0] = v.x; dst[1] = v.y; dst[2] = v.z; dst[3] = v.w;
        }
#endif
        for (int idx = tid; idx < (FP - F) * DD; idx += 128)
            sX[(F + idx / DD) * XS + (idx % DD)] = 0.0f;
        for (int idx = tid; idx < DD * 16; idx += 128) {
            int d = idx >> 4, n = idx & 15;
            sWq[idx] = wq[d * 64 + n];
            sWk[idx] = wk[d * 64 + n];
        }
        if (tid < DD) sWv[tid] = wv[tid * 64];
#ifdef HAVE_ASYNC_LDS
        async_wait0();
#endif
    }
    __syncthreads();

    // ---- v column (head0, dim0): plain VALU dot products ------------------
    if (tid < F) {
        float acc = 0.0f;
        #pragma unroll 8
        for (int d = 0; d < DD; ++d) acc += sX[tid * XS + d] * sWv[d];
        sVcol[tid] = acc;
    }

    // ---- Q and K GEMMs via f32 WMMA (scalar-uniform job loop) -------------
    for (int job = wave; job < 14; job += 4) {
        const int m = job >> 1;
        const float* W  = (job & 1) ? sWk : sWq;
        float*       Om = (job & 1) ? sK  : sQ;
        const int mbase = m * 16;
        v8f acc = {0.f, 0.f, 0.f, 0.f, 0.f, 0.f, 0.f, 0.f};
        #pragma unroll
        for (int kc = 0; kc < DD; kc += 4) {
            v2f a, bm;
            a.x  = sX[(mbase + lrow) * XS + kc + 2 * half];
            a.y  = sX[(mbase + lrow) * XS + kc + 2 * half + 1];
            bm.x = W[(kc + 2 * half) * 16 + lrow];
            bm.y = W[(kc + 2 * half + 1) * 16 + lrow];
            acc = wmma_f32(a, bm, acc);
        }
        #pragma unroll
        for (int r = 0; r < 8; ++r)
            Om[(mbase + r + 8 * half) * QS + lrow] = acc[r];
    }
    __syncthreads();

    // ---- S = Q K^T / 4, masked softmax, z = A * vcol ----------------------
    for (int m = wave; m < 7; m += 4) {
        const int mbase = m * 16;
        v8f sacc[7];
        #pragma unroll
        for (int n = 0; n < 7; ++n)
            sacc[n] = (v8f){0.f, 0.f, 0.f, 0.f, 0.f, 0.f, 0.f, 0.f};

        #pragma unroll
        for (int kc = 0; kc < 16; kc += 4) {
            v2f a;
            a.x = sQ[(mbase + lrow) * QS + kc + 2 * half];
            a.y = sQ[(mbase + lrow) * QS + kc + 2 * half + 1];
            #pragma unroll
            for (int n = 0; n < 7; ++n) {
                v2f bt;  // B = K^T chunk: B[k][n] = K[nbase+n][kc+k]
                bt.x = sK[(n * 16 + lrow) * QS + kc + 2 * half];
                bt.y = sK[(n * 16 + lrow) * QS + kc + 2 * half + 1];
                sacc[n] = wmma_f32(a, bt, sacc[n]);
            }
        }

        // row-wise masked softmax; C/D layout: VGPR r -> rows r / r+8, col = n*16+lrow
        #pragma unroll
        for (int r = 0; r < 8; ++r) {
            const int row = mbase + r + 8 * half;
            float mx = -3.402823466e38f;
            #pragma unroll
            for (int n = 0; n < 7; ++n) {
                int c = n * 16 + lrow;
                float s = sacc[n][r] * 0.25f;
                if (c < F) mx = fmaxf(mx, s);
            }
            mx = fmaxf(mx, __shfl_xor(mx, 1, 32));
            mx = fmaxf(mx, __shfl_xor(mx, 2, 32));
            mx = fmaxf(mx, __shfl_xor(mx, 4, 32));
            mx = fmaxf(mx, __shfl_xor(mx, 8, 32));

            float sum = 0.0f, za = 0.0f;
            #pragma unroll
            for (int n = 0; n < 7; ++n) {
                int c = n * 16 + lrow;
                if (c < F) {
                    float p = __expf(sacc[n][r] * 0.25f - mx);
                    sum += p;
                    za  += p * sVcol[c];
                }
            }
            sum += __shfl_xor(sum, 1, 32);
            sum += __shfl_xor(sum, 2, 32);
            sum += __shfl_xor(sum, 4, 32);
            sum += __shfl_xor(sum, 8, 32);
            za  += __shfl_xor(za, 1, 32);
            za  += __shfl_xor(za, 2, 32);
            za  += __shfl_xor(za, 4, 32);
            za  += __shfl_xor(za, 8, 32);

            if (lrow == 0) sZ[row] = za / sum;
        }
    }
    __syncthreads();

    // ---- pair expansion, fully coalesced flat loop ------------------------
    // p -> (i, j): i = floor((199 - sqrt(39601 - 8p)) / 2), exact at tile
    // boundaries in fp32 (all values < 2^24); fix-up loops for safety.
    float* outb = out + (size_t)b * NPAIR;
    for (int p = tid; p < NPAIR; p += 128) {
        float disc = 39601.0f - 8.0f * (float)p;
        int i = (int)((199.0f - sqrtf(disc)) * 0.5f);
        if (i > F - 2) i = F - 2;
        while (i > 0 && (F - 1) * i - ((i * (i - 1)) >> 1) > p) --i;
        while ((F - 1) * (i + 1) - (((i + 1) * i) >> 1) <= p) ++i;
        int off = (F - 1) * i - ((i * (i - 1)) >> 1);
        int j = i + 1 + (p - off);
        outb[p] = sZ[i] * sX[j * XS];
    }
}

extern "C" void kernel_launch(void* const* d_in, const int* in_sizes, int n_in,
                              void* d_out, int out_size, void* d_ws, size_t ws_size,
                              hipStream_t stream) {
    const float* x  = (const float*)d_in[0];
    const float* wq = (const float*)d_in[1];
    const float* wk = (const float*)d_in[2];
    const float* wv = (const float*)d_in[3];
    float* out = (float*)d_out;
    dcap_attn_pairs_kernel<<<BATCH, 128, 0, stream>>>(x, wq, wk, wv, out);
}